// Cross_13718125543491
// MI455X (gfx1250) — compile-verified
//
#include <hip/hip_runtime.h>
#include <stdint.h>

// ---------------------------------------------------------------------------
// out[b,f] = x0[b,f] * dot(x[b,:], w) + bias[f] + x[b,f]
// B=16384 rows, F=2048 cols, fp32. Bandwidth-bound: 384 MB => ~16.5us @ 23.3TB/s.
// CDNA5 paths used: TENSOR_LOAD_TO_LDS (TDM, iterate mode, padded LDS stride),
// V_WMMA_F32_16X16X4_F32 for the row-dot, s_wait_tensorcnt, global_prefetch_b8,
// non-temporal load/store for the streaming operands.
// ---------------------------------------------------------------------------

typedef __attribute__((ext_vector_type(2))) float v2f;
typedef __attribute__((ext_vector_type(4))) float v4f;
typedef __attribute__((ext_vector_type(8))) float v8f;
typedef __attribute__((ext_vector_type(4))) unsigned int v4u;
typedef __attribute__((ext_vector_type(4))) int v4i;
typedef __attribute__((ext_vector_type(8))) int v8i;

static constexpr int kB      = 16384;
static constexpr int kF      = 2048;
static constexpr int kRows   = 16;          // rows per block (WMMA M)
static constexpr int kChunk  = 512;         // K columns staged per TDM round
static constexpr int kStride = kChunk + 4;  // 516 dwords; 516 % 64 == 4 => conflict-free A gather
static constexpr int kNChunks = kF / kChunk;

// LDS layout (dword offsets)
static constexpr int X_OFF   = 0;                        // 16 * 516 = 8256 dwords
static constexpr int W_OFF   = kRows * kStride;          // 8256, 2048 dwords (full weights)
static constexpr int RED_OFF = W_OFF + kF;               // 10304, 8 waves * 16 partials
static constexpr int SV_OFF  = RED_OFF + 128;            // 10432, 16 row sums
static constexpr int LDS_DW  = SV_OFF + 16;              // 10448 dwords = 41792 bytes

// Issue one TDM load: `rows` rows of `cols` fp32, global row stride g_stride
// elements, LDS row stride lds_stride elements (iterate mode: 1 row per
// iteration so LDS gets the padded stride).
__device__ __forceinline__ void tdm_rows_to_lds(unsigned lds_byte, const float* g,
                                                int cols, int rows,
                                                int lds_stride, int g_stride) {
  unsigned long long ga = (unsigned long long)(uintptr_t)g;
  // Group 0: count=1 | lds_addr | global_addr[56:0] | type=2 at bits 127:126
  v4u g0 = { 1u,
             lds_byte,
             (unsigned)ga,
             (unsigned)((ga >> 32) & 0x01FFFFFFull) | 0x80000000u };
  // Group 1: data_size=2 (4B) at [17:16], iterate_enable at [19],
  // tensor_dim0 [79:48], tensor_dim1 [111:80], tile_dim0 [127:112],
  // tile_dim1 [143:128]=1, tensor_dim0_stride [207:160]
  v8i g1 = { (int)((2u << 16) | (1u << 19)),
             (int)((unsigned)cols << 16),
             (int)((((unsigned)cols >> 16) & 0xFFFFu) | ((unsigned)rows << 16)),
             (int)((unsigned)cols << 16),
             1,
             g_stride,
             0, 0 };
  // Group 2 (iterate mode): tensor_dim2=0, lds_addr_increment,
  // global_addr_increment [111:64], iterate_count [127:112]
  v4i g2 = { 0, lds_stride, g_stride, (rows - 1) << 16 };
  v4i g3 = { 0, 0, 0, 0 };
  v8i gz = { 0, 0, 0, 0, 0, 0, 0, 0 };   // prod-lane 6-arg form: extra group word
  __builtin_amdgcn_tensor_load_to_lds(g0, g1, g2, g3, gz, 0);
}

__global__ __launch_bounds__(256)
void cross_fused_kernel(const float* __restrict__ x0, const float* __restrict__ x,
                        const float* __restrict__ w, const float* __restrict__ bias,
                        float* __restrict__ out) {
  extern __shared__ float smem[];
  const int t    = threadIdx.x;
  const int lane = t & 31;
  const int wid  = t >> 5;                 // 8 waves
  const int bid  = blockIdx.x;             // 1024 blocks, 16 rows each

  const unsigned lds_base = (unsigned)(uintptr_t)(void*)smem;  // flat LDS addr low 32 = LDS offset
  const float* xg = x + (size_t)bid * kRows * kF;

  // WMMA A-operand lane mapping for 16x16x4 f32:
  //   lanes 0-15  hold (M=lane,   K = k0, k0+1) in v0,v1
  //   lanes 16-31 hold (M=lane-16,K = k0+2,k0+3)
  const int m    = lane & 15;
  const int koff = (lane >> 4) << 1;

  v8f acc = (v8f)0.0f;

  #pragma unroll 1
  for (int chunk = 0; chunk < kNChunks; ++chunk) {
    if (chunk == 0) {
      // TDM for chunk 0 overlapped with weights staging + x0 prefetch.
      if (wid == 0)
        tdm_rows_to_lds(lds_base + X_OFF * 4u, xg, kChunk, kRows, kStride, kF);
      { // weights -> LDS, coalesced float4 (2048 floats = 512 float4)
        const v4f* w4 = (const v4f*)w;
        v4f* wl4 = (v4f*)(smem + W_OFF);
        wl4[t]       = w4[t];
        wl4[t + 256] = w4[t + 256];
      }
      { // warm L2 with this block's x0 tile (128 KB): 256 thr * 2 * 256B
        const char* p0 = (const char*)x0 + (size_t)bid * kRows * kF * 4;
        __builtin_prefetch(p0 + t * 512, 0, 1);
        __builtin_prefetch(p0 + t * 512 + 256, 0, 1);
      }
      if (wid == 0) __builtin_amdgcn_s_wait_tensorcnt(0);
      __syncthreads();
    } else {
      __syncthreads();   // all waves done reading previous chunk
      if (wid == 0) {
        tdm_rows_to_lds(lds_base + X_OFF * 4u, xg + chunk * kChunk,
                        kChunk, kRows, kStride, kF);
        __builtin_amdgcn_s_wait_tensorcnt(0);
      }
      __syncthreads();
    }

    // Each wave owns a 64-wide K slice of this 512-wide chunk: 16 WMMAs.
    const int cbase = wid * 64;
    #pragma unroll
    for (int s = 0; s < 16; ++s) {
      const int c = cbase + s * 4;
      // A: conflict-free ds_load_b64 (stride 516 => bank = 4*m + c (+0..3))
      const v2f a = *(const v2f*)(smem + X_OFF + m * kStride + c + koff);
      // B: weights broadcast across all 16 columns (same K pairing as A)
      const v2f b = *(const v2f*)(smem + W_OFF + chunk * kChunk + c + koff);
      acc = __builtin_amdgcn_wmma_f32_16x16x4_f32(
          /*neg_a=*/false, a, /*neg_b=*/false, b,
          /*c_mod=*/(short)0, acc, /*reuse_a=*/false, /*reuse_b=*/false);
    }
  }

  // Every column of D is the same partial dot. Lane 0 holds M=0..7 in v0..7,
  // lane 16 holds M=8..15. Reduce the 8 per-wave partials through LDS.
  if (lane == 0) {
    #pragma unroll
    for (int i = 0; i < 8; ++i) smem[RED_OFF + wid * 16 + i] = acc[i];
  } else if (lane == 16) {
    #pragma unroll
    for (int i = 0; i < 8; ++i) smem[RED_OFF + wid * 16 + 8 + i] = acc[i];
  }
  __syncthreads();
  if (t < 16) {
    float s = 0.0f;
    #pragma unroll
    for (int wv = 0; wv < 8; ++wv) s += smem[RED_OFF + wv * 16 + t];
    smem[SV_OFF + t] = s;
  }
  __syncthreads();

  // Phase 2: fully coalesced float4 streams.
  //  - x   : regular load, hits L2 (TDM just pulled it through GL2)
  //  - x0  : non-temporal load  (never reused; keep L2 for x)
  //  - out : non-temporal store (never reused)
  const v4f* x4g  = (const v4f*)x;
  const v4f* x04g = (const v4f*)x0;
  const v4f* b4g  = (const v4f*)bias;
  v4f*       o4g  = (v4f*)out;
  const int g4base = bid * (kRows * kF / 4);   // 8192 float4 per block

  #pragma unroll 4
  for (int j = 0; j < 32; ++j) {
    const int f = j * 256 + t;                 // 0..8191
    const v4f xv  = x4g[g4base + f];
    const v4f x0v = __builtin_nontemporal_load(&x04g[g4base + f]);
    const v4f bv  = b4g[f & 511];
    const float s = smem[SV_OFF + (f >> 9)];
    const v4f o = x0v * s + bv + xv;
    __builtin_nontemporal_store(o, &o4g[g4base + f]);
  }
}

extern "C" void kernel_launch(void* const* d_in, const int* in_sizes, int n_in,
                              void* d_out, int out_size, void* d_ws, size_t ws_size,
                              hipStream_t stream) {
  (void)in_sizes; (void)n_in; (void)out_size; (void)d_ws; (void)ws_size;
  const float* x0   = (const float*)d_in[0];
  const float* x    = (const float*)d_in[1];
  const float* w    = (const float*)d_in[2];
  const float* bias = (const float*)d_in[3];
  float* out = (float*)d_out;

  dim3 grid(kB / kRows);   // 1024 blocks
  dim3 block(256);         // 8 wave32 per block
  cross_fused_kernel<<<grid, block, LDS_DW * sizeof(float), stream>>>(
      x0, x, w, bias, out);
}